// DequantSiluAndMulQuant_56908316672719
// MI455X (gfx1250) — compile-verified
//
#include <hip/hip_runtime.h>
#include <stdint.h>

// ---------------------------------------------------------------------------
// DequantSiluAndMulQuant for MI455X (gfx1250)
//
// Memory-bound streaming op (~1.08 GB traffic -> ~46us floor @ 23.3 TB/s).
// No matmul => WMMA not applicable. CDNA5 path: Tensor Data Mover
// (tensor_load_to_lds + s_wait_tensorcnt) DMAs each 88KB row into LDS;
// compute is 4-wide (b128 LDS ops), fast exp/rcp, NT b128 stores.
// ---------------------------------------------------------------------------

#define NUM_TOKENS 8192
#define TWO_D      22016
#define HALF_D     11008
#define NVEC       (HALF_D / 4)   // 2752 float4 per half-row
#define THREADS    256
#define WAVES      (THREADS / 32)

typedef unsigned int v4u __attribute__((ext_vector_type(4)));
typedef int          v8i __attribute__((ext_vector_type(8)));
typedef int          v4i __attribute__((ext_vector_type(4)));
typedef float        v4f __attribute__((ext_vector_type(4)));

// Byte offset of a shared-memory object inside the LDS allocation
// (generic pointer -> addrspace(3) -> integer). Device-only construct.
static __device__ __forceinline__ unsigned lds_byte_offset(void* p) {
#if defined(__HIP_DEVICE_COMPILE__)
    return (unsigned)(unsigned long long)(__attribute__((address_space(3))) void*)p;
#else
    (void)p;
    return 0;
#endif
}

#if defined(__HIP_DEVICE_COMPILE__) && __has_builtin(__builtin_amdgcn_tensor_load_to_lds)
#define USE_TDM 1
#else
#define USE_TDM 0
#endif

#if USE_TDM
// Issue one TDM descriptor: contiguous 1-D tile of `nelem` int32 starting at
// `gptr` (global) into LDS byte offset `lds_off`. D# layout per CDNA5 ISA ch.8.
static __device__ __forceinline__ void tdm_load_row_1d(const int* gptr,
                                                       unsigned lds_off,
                                                       int nelem) {
    uint64_t ga = (uint64_t)gptr;
    // Group 0: [1:0]=count=1 (valid user desc), [63:32]=lds_addr,
    //          [120:64]=global_addr, [127:126]=type=2 ("image")
    v4u g0 = { 1u,
               lds_off,
               (unsigned)(ga & 0xFFFFFFFFu),
               (unsigned)((ga >> 32) & 0x01FFFFFFu) | 0x80000000u };
    // Group 1:
    //  d0: workgroup_mask=0 (not in cluster), data_size=2 (4B) at [17:16]
    //  d1: [63:48] = tensor_dim0[15:0]
    //  d2: [79:64] = tensor_dim0[31:16]=0 ; [95:80] tensor_dim1[15:0] = 1
    //  d3: [111:96] tensor_dim1[31:16]=0 ; [127:112] tile_dim0
    //  d4: tile_dim1 = 1 ; tile_dim2 = 0
    //  d5/d6/d7: tensor_dim0_stride = nelem ; tensor_dim1_stride = 0
    v8i g1 = { (int)(2u << 16),
               (int)((unsigned)(nelem & 0xFFFF) << 16),
               (int)(1u << 16),
               (int)((unsigned)(nelem & 0xFFFF) << 16),
               1,
               nelem,
               0,
               0 };
    v4i z4 = { 0, 0, 0, 0 };
    v8i z8 = { 0, 0, 0, 0, 0, 0, 0, 0 };
    // 6-arg form on this toolchain: (g0, g1, g2, g3, g4, cpol)
    __builtin_amdgcn_tensor_load_to_lds(g0, g1, z4, z4, z8, 0);
}
#endif

// silu(g) * u with fast native ops: ~1ulp error, far below int8 quant step.
static __device__ __forceinline__ float silu_mul(float g, float u) {
#if defined(__HIP_DEVICE_COMPILE__)
    float e   = __expf(-g);                       // v_exp_f32 path
    float sig = __builtin_amdgcn_rcpf(1.0f + e);  // v_rcp_f32
    return g * sig * u;
#else
    return (g / (1.0f + expf(-g))) * u;
#endif
}

__global__ __launch_bounds__(THREADS)
void DequantSiluAndMulQuant_56908316672719_kernel(const int* __restrict__ x,
                                                  const float* __restrict__ dscale_p,
                                                  float* __restrict__ out_q,
                                                  float* __restrict__ out_scale) {
    __shared__ __align__(16) int smem[TWO_D];  // row staging; gate half reused for tmp
    __shared__ float wred[WAVES];
    __shared__ float s_scale;

    const int row = blockIdx.x;
    const int tid = threadIdx.x;
    const int* rowp = x + (size_t)row * TWO_D;

    // ---- Stage the full row (gate + up halves) into LDS via TDM -----------
#if USE_TDM
    if (tid < 32) {  // wave 0 only: TDM is per-wave, EXEC-independent
        tdm_load_row_1d(rowp, lds_byte_offset(smem), TWO_D);
        __builtin_amdgcn_s_wait_tensorcnt(0);
    }
#elif defined(__HIP_DEVICE_COMPILE__)
    {
        const int nv = TWO_D / 4;  // 5504 x b128
        for (int v = tid; v < nv; v += THREADS) {
            unsigned lo = lds_byte_offset(smem) + (unsigned)v * 16u;
            const int* ga = rowp + v * 4;
            asm volatile("global_load_async_to_lds_b128 %0, %1, off"
                         :: "v"(lo), "v"(ga)
                         : "memory");
        }
#if __has_builtin(__builtin_amdgcn_s_wait_asynccnt)
        __builtin_amdgcn_s_wait_asynccnt(0);
#else
        asm volatile("s_wait_asynccnt 0" ::: "memory");
#endif
    }
#endif
    __syncthreads();

    // ---- Pass 1: tmp = silu(gate)*up in-place (b128 LDS ops); row max -----
    const float dq = *dscale_p;
    const v4i*  sg4 = (const v4i*)smem;          // gate half, vec4
    const v4i*  su4 = (const v4i*)smem + NVEC;   // up half, vec4
    v4f*        tf4 = (v4f*)smem;                // tmp overwrites gate half
    float m = 0.0f;
    for (int v = tid; v < NVEC; v += THREADS) {
        v4i gi = sg4[v];
        v4i ui = su4[v];
        v4f t;
        t.x = silu_mul((float)gi.x * dq, (float)ui.x * dq);
        t.y = silu_mul((float)gi.y * dq, (float)ui.y * dq);
        t.z = silu_mul((float)gi.z * dq, (float)ui.z * dq);
        t.w = silu_mul((float)gi.w * dq, (float)ui.w * dq);
        tf4[v] = t;
        m = fmaxf(m, fmaxf(fmaxf(fabsf(t.x), fabsf(t.y)),
                           fmaxf(fabsf(t.z), fabsf(t.w))));
    }

    // ---- Row-max reduction: wave32 shuffle tree, then across 8 waves ------
    for (int off = 16; off > 0; off >>= 1)
        m = fmaxf(m, __shfl_xor(m, off, 32));
    if ((tid & 31) == 0) wred[tid >> 5] = m;
    __syncthreads();
    if (tid == 0) {
        float mm = wred[0];
        #pragma unroll
        for (int w = 1; w < WAVES; ++w) mm = fmaxf(mm, wred[w]);
        float sc = mm / 127.0f;
        s_scale = sc;
        out_scale[row] = sc;
    }
    __syncthreads();

    // ---- Pass 2: quantize tmp from LDS, NT b128 stores of q-as-float ------
    const float sc  = s_scale;
    const float inv = 1.0f / sc;                   // one divide per thread
    v4f* oq4 = (v4f*)(out_q + (size_t)row * HALF_D);
    for (int v = tid; v < NVEC; v += THREADS) {
        v4f t = tf4[v];
        v4f q;
        q.x = fminf(fmaxf(rintf(t.x * inv), -128.0f), 127.0f);
        q.y = fminf(fmaxf(rintf(t.y * inv), -128.0f), 127.0f);
        q.z = fminf(fmaxf(rintf(t.z * inv), -128.0f), 127.0f);
        q.w = fminf(fmaxf(rintf(t.w * inv), -128.0f), 127.0f);
        __builtin_nontemporal_store(q, &oq4[v]);   // stream past L2
    }
}

extern "C" void kernel_launch(void* const* d_in, const int* in_sizes, int n_in,
                              void* d_out, int out_size, void* d_ws, size_t ws_size,
                              hipStream_t stream) {
    (void)in_sizes; (void)n_in; (void)out_size; (void)d_ws; (void)ws_size;
    const int*   x   = (const int*)d_in[0];
    const float* dsc = (const float*)d_in[1];
    float* out_q     = (float*)d_out;
    float* out_scale = out_q + (size_t)NUM_TOKENS * HALF_D;

    DequantSiluAndMulQuant_56908316672719_kernel
        <<<NUM_TOKENS, THREADS, 0, stream>>>(x, dsc, out_q, out_scale);
}